// Deform_47742856462448
// MI455X (gfx1250) — compile-verified
//
#include <hip/hip_runtime.h>
#include <hip/hip_bf16.h>

#define Bn 2
#define Cn 64
#define Hn 128
#define Wn 128
#define HWn (Hn * Wn)

typedef float v2f __attribute__((ext_vector_type(2)));
typedef float v8f __attribute__((ext_vector_type(8)));

// ---------------------------------------------------------------------------
// Kernel 1: s = BN1(dw1x15(x)+b1) + BN2(dw15x1(x)+b2) + BN3(dw3x3(x)+b3)
// One thread per output element; all taps are L2/L1-hot (x = 8MB).
// ---------------------------------------------------------------------------
__global__ void k1_branch(const float* __restrict__ x,
                          const float* __restrict__ w1, const float* __restrict__ b1,
                          const float* __restrict__ s1, const float* __restrict__ bb1,
                          const float* __restrict__ w2, const float* __restrict__ b2,
                          const float* __restrict__ s2, const float* __restrict__ bb2,
                          const float* __restrict__ w3, const float* __restrict__ b3,
                          const float* __restrict__ s3, const float* __restrict__ bb3,
                          float* __restrict__ sOut)
{
    int idx = blockIdx.x * blockDim.x + threadIdx.x;
    int w = idx & (Wn - 1);
    int h = (idx >> 7) & (Hn - 1);
    int c = (idx >> 14) & (Cn - 1);
    int b = idx >> 20;
    const float* xb = x + (size_t)(b * Cn + c) * HWn;

    float a1 = 0.f;
#pragma unroll
    for (int j = 0; j < 15; ++j) {
        int ww = w - 7 + j;
        float v = (ww >= 0 && ww < Wn) ? xb[h * Wn + ww] : 0.f;
        a1 = fmaf(v, w1[c * 15 + j], a1);
    }
    float a2 = 0.f;
#pragma unroll
    for (int i = 0; i < 15; ++i) {
        int hh = h - 7 + i;
        float v = (hh >= 0 && hh < Hn) ? xb[hh * Wn + w] : 0.f;
        a2 = fmaf(v, w2[c * 15 + i], a2);
    }
    float a3 = 0.f;
#pragma unroll
    for (int i = 0; i < 3; ++i)
#pragma unroll
        for (int j = 0; j < 3; ++j) {
            int hh = h - 1 + i, ww = w - 1 + j;
            float v = (hh >= 0 && hh < Hn && ww >= 0 && ww < Wn) ? xb[hh * Wn + ww] : 0.f;
            a3 = fmaf(v, w3[c * 9 + i * 3 + j], a3);
        }

    float r = s1[c] * (a1 + b1[c]) + bb1[c]
            + s2[c] * (a2 + b2[c]) + bb2[c]
            + s3[c] * (a3 + b3[c]) + bb3[c];
    sOut[idx] = r;
}

// ---------------------------------------------------------------------------
// Kernel 2: off[18, BHW] = BN4(balW[18,64] . s[64, BHW] + bal_b)
// V_WMMA_F32_16X16X4_F32, one wave = 16x16 (M x pixels) tile, K=64.
// A rows >= 18: clamped address + 0-mask (no exec divergence).
// All A/B fragments preloaded -> one load clause, then 16 WMMAs back-to-back.
// ---------------------------------------------------------------------------
__global__ void k2_bal(const float* __restrict__ sIn, const float* __restrict__ balW,
                       const float* __restrict__ balB, const float* __restrict__ bn4s,
                       const float* __restrict__ bn4b, float* __restrict__ offOut)
{
    int wave = (blockIdx.x * blockDim.x + threadIdx.x) >> 5;
    int lane = threadIdx.x & 31;
    int half = lane >> 4;      // 0: K {0,1}; 1: K {2,3}
    int l    = lane & 15;

    const int NT = HWn / 16;   // 1024 pixel tiles per image
    int b   = wave / (2 * NT);
    int rem = wave % (2 * NT);
    int mt  = rem / NT;
    int nt  = rem % NT;
    int mbase = mt * 16;
    int pbase = nt * 16;

    int   ma    = mbase + l;                 // A-matrix row for this lane
    float amask = (ma < 18) ? 1.f : 0.f;
    int   mac   = (ma < 18) ? ma : 17;       // clamped (always-valid) row

    const v2f*   aRow  = (const v2f*)(balW + mac * 64 + half * 2);      // 8B aligned
    const float* bBase = sIn + (size_t)b * Cn * HWn
                       + (size_t)(half * 2) * HWn + pbase + l;

    v2f av[16];
    v2f bv[16];
#pragma unroll
    for (int kk = 0; kk < 16; ++kk) {
        av[kk] = aRow[kk * 2];               // global_load_b64
        av[kk].x *= amask;
        av[kk].y *= amask;
        bv[kk].x = bBase[(size_t)(kk * 4) * HWn];
        bv[kk].y = bBase[(size_t)(kk * 4 + 1) * HWn];
    }

    v8f acc = {};
#pragma unroll
    for (int kk = 0; kk < 16; ++kk) {
        acc = __builtin_amdgcn_wmma_f32_16x16x4_f32(false, av[kk], false, bv[kk],
                                                    (short)0, acc, false, false);
    }

    float* ob = offOut + (size_t)b * 18 * HWn + pbase + l;
#pragma unroll
    for (int r = 0; r < 8; ++r) {
        int m = mbase + half * 8 + r;
        if (m < 18) {
            ob[m * HWn] = acc[r] * bn4s[m] + (bn4s[m] * balB[m] + bn4b[m]);
        }
    }
}

// ---------------------------------------------------------------------------
// Kernel 3: deformable depthwise sampling, 4 dilations {2,3,4,5}, 9 taps,
// bilinear. One thread per pixel; 18 offsets hoisted to registers; dw weights
// for all 4 dilations staged in LDS; channels tiled x8 so each (d,k) position
// computation amortizes over 8 channel FMAs. Gathers are L2-resident (x=8MB).
// ---------------------------------------------------------------------------
__global__ __launch_bounds__(128)
void k3_deform(const float* __restrict__ x, const float* __restrict__ off,
               const float* __restrict__ dw2, const float* __restrict__ dw3,
               const float* __restrict__ dw4, const float* __restrict__ dw5,
               float* __restrict__ dout)
{
    __shared__ float dwl[4 * Cn * 9];
    for (int t = threadIdx.x; t < 4 * Cn * 9; t += blockDim.x) {
        int q = t / (Cn * 9), r = t % (Cn * 9);
        const float* src = (q == 0) ? dw2 : (q == 1) ? dw3 : (q == 2) ? dw4 : dw5;
        dwl[t] = src[r];
    }
    __syncthreads();

    int gp = blockIdx.x * blockDim.x + threadIdx.x;
    int b = gp >> 14;
    int p = gp & (HWn - 1);
    int h = p >> 7;
    int w = p & (Wn - 1);

    const float* offB = off + (size_t)b * 18 * HWn + p;
    float offv[18];
#pragma unroll
    for (int j = 0; j < 18; ++j) offv[j] = offB[j * HWn];

    const float* xB = x + (size_t)b * Cn * HWn;

    for (int ct = 0; ct < 8; ++ct) {
        float acc[8] = {0.f, 0.f, 0.f, 0.f, 0.f, 0.f, 0.f, 0.f};
        for (int k = 0; k < 9; ++k) {
            float offy = offv[2 * k], offx = offv[2 * k + 1];
            int ky = k / 3 - 1, kx = k % 3 - 1;
#pragma unroll
            for (int di = 0; di < 4; ++di) {
                int dil = di + 2;
                float py = offy + (float)(h + dil * ky);
                float px = offx + (float)(w + dil * kx);
                float y0 = floorf(py), x0 = floorf(px);
                float wy = py - y0, wx = px - x0;
                float y1 = y0 + 1.f, x1 = x0 + 1.f;
                bool vy0 = (y0 >= 0.f) && (y0 <= (float)(Hn - 1));
                bool vy1 = (y1 >= 0.f) && (y1 <= (float)(Hn - 1));
                bool vx0 = (x0 >= 0.f) && (x0 <= (float)(Wn - 1));
                bool vx1 = (x1 >= 0.f) && (x1 <= (float)(Wn - 1));
                int iy0 = (int)fminf(fmaxf(y0, 0.f), (float)(Hn - 1));
                int iy1 = (int)fminf(fmaxf(y1, 0.f), (float)(Hn - 1));
                int ix0 = (int)fminf(fmaxf(x0, 0.f), (float)(Wn - 1));
                int ix1 = (int)fminf(fmaxf(x1, 0.f), (float)(Wn - 1));
                float w00 = (vy0 && vx0) ? (1.f - wy) * (1.f - wx) : 0.f;
                float w01 = (vy0 && vx1) ? (1.f - wy) * wx : 0.f;
                float w10 = (vy1 && vx0) ? wy * (1.f - wx) : 0.f;
                float w11 = (vy1 && vx1) ? wy * wx : 0.f;
                int i00 = iy0 * Wn + ix0, i01 = iy0 * Wn + ix1;
                int i10 = iy1 * Wn + ix0, i11 = iy1 * Wn + ix1;
#pragma unroll
                for (int ci = 0; ci < 8; ++ci) {
                    int c = ct * 8 + ci;
                    const float* xc = xB + (size_t)c * HWn;
                    float v = w00 * xc[i00] + w01 * xc[i01]
                            + w10 * xc[i10] + w11 * xc[i11];
                    acc[ci] = fmaf(v, dwl[di * (Cn * 9) + c * 9 + k], acc[ci]);
                }
            }
        }
#pragma unroll
        for (int ci = 0; ci < 8; ++ci)
            dout[(size_t)(b * Cn + ct * 8 + ci) * HWn + p] = acc[ci];
    }
}

// ---------------------------------------------------------------------------
// Kernel 4: out = (BN5(finW[64,64] . d[64, BHW] + fin_b)) * x
// Same WMMA tiling as k2 (M = 64 -> 4 tiles of 16); preloaded fragments,
// fused bias + BN + elementwise multiply with x in the epilogue.
// ---------------------------------------------------------------------------
__global__ void k4_final(const float* __restrict__ dIn, const float* __restrict__ finW,
                         const float* __restrict__ finB, const float* __restrict__ bn5s,
                         const float* __restrict__ bn5b, const float* __restrict__ x,
                         float* __restrict__ out)
{
    int wave = (blockIdx.x * blockDim.x + threadIdx.x) >> 5;
    int lane = threadIdx.x & 31;
    int half = lane >> 4;
    int l    = lane & 15;

    const int NT = HWn / 16;
    int b   = wave / (4 * NT);
    int rem = wave % (4 * NT);
    int mt  = rem / NT;
    int nt  = rem % NT;
    int mbase = mt * 16;
    int pbase = nt * 16;

    int ma = mbase + l;
    const v2f*   aRow  = (const v2f*)(finW + ma * 64 + half * 2);       // 8B aligned
    const float* bBase = dIn + (size_t)b * Cn * HWn
                       + (size_t)(half * 2) * HWn + pbase + l;

    v2f av[16];
    v2f bv[16];
#pragma unroll
    for (int kk = 0; kk < 16; ++kk) {
        av[kk] = aRow[kk * 2];               // global_load_b64
        bv[kk].x = bBase[(size_t)(kk * 4) * HWn];
        bv[kk].y = bBase[(size_t)(kk * 4 + 1) * HWn];
    }

    v8f acc = {};
#pragma unroll
    for (int kk = 0; kk < 16; ++kk) {
        acc = __builtin_amdgcn_wmma_f32_16x16x4_f32(false, av[kk], false, bv[kk],
                                                    (short)0, acc, false, false);
    }

#pragma unroll
    for (int r = 0; r < 8; ++r) {
        int m = mbase + half * 8 + r;
        size_t o = (size_t)(b * Cn + m) * HWn + pbase + l;
        float v = (acc[r] + finB[m]) * bn5s[m] + bn5b[m];
        out[o] = v * x[o];
    }
}

// ---------------------------------------------------------------------------
extern "C" void kernel_launch(void* const* d_in, const int* in_sizes, int n_in,
                              void* d_out, int out_size, void* d_ws, size_t ws_size,
                              hipStream_t stream)
{
    const float* x     = (const float*)d_in[0];
    const float* off1w = (const float*)d_in[1];
    const float* off1b = (const float*)d_in[2];
    const float* bn1s  = (const float*)d_in[3];
    const float* bn1b  = (const float*)d_in[4];
    const float* off2w = (const float*)d_in[5];
    const float* off2b = (const float*)d_in[6];
    const float* bn2s  = (const float*)d_in[7];
    const float* bn2b  = (const float*)d_in[8];
    const float* off3w = (const float*)d_in[9];
    const float* off3b = (const float*)d_in[10];
    const float* bn3s  = (const float*)d_in[11];
    const float* bn3b  = (const float*)d_in[12];
    const float* balw  = (const float*)d_in[13];
    const float* balb  = (const float*)d_in[14];
    const float* bn4s  = (const float*)d_in[15];
    const float* bn4b  = (const float*)d_in[16];
    const float* dw2   = (const float*)d_in[17];
    const float* dw3   = (const float*)d_in[18];
    const float* dw4   = (const float*)d_in[19];
    const float* dw5   = (const float*)d_in[20];
    const float* finw  = (const float*)d_in[21];
    const float* finb  = (const float*)d_in[22];
    const float* bn5s  = (const float*)d_in[23];
    const float* bn5b  = (const float*)d_in[24];

    float* ws    = (float*)d_ws;
    float* sWs   = ws;                                   // B*C*HW   = 2,097,152 f
    float* offWs = sWs + (size_t)Bn * Cn * HWn;          // B*18*HW  =   589,824 f
    float* dWs   = offWs + (size_t)Bn * 18 * HWn;        // B*C*HW   = 2,097,152 f

    // 1. branch depthwise convs + BN -> s
    k1_branch<<<(Bn * Cn * HWn) / 256, 256, 0, stream>>>(
        x, off1w, off1b, bn1s, bn1b, off2w, off2b, bn2s, bn2b,
        off3w, off3b, bn3s, bn3b, sWs);

    // 2. 1x1 conv 64->18 (WMMA f32) + BN -> offsets
    {
        int waves = Bn * 2 * (HWn / 16);   // 4096
        k2_bal<<<(waves * 32) / 256, 256, 0, stream>>>(
            sWs, balw, balb, bn4s, bn4b, offWs);
    }

    // 3. deformable sampling, 4 dilations summed -> dWs
    k3_deform<<<(Bn * HWn) / 128, 128, 0, stream>>>(
        x, offWs, dw2, dw3, dw4, dw5, dWs);

    // 4. 1x1 conv 64->64 (WMMA f32) + BN, * x -> out
    {
        int waves = Bn * 4 * (HWn / 16);   // 8192
        k4_final<<<(waves * 32) / 256, 256, 0, stream>>>(
            dWs, finw, finb, bn5s, bn5b, x, (float*)d_out);
    }
}